// GIN_10350871184011
// MI455X (gfx1250) — compile-verified
//
#include <hip/hip_runtime.h>

typedef float v2f __attribute__((ext_vector_type(2)));
typedef float v8f __attribute__((ext_vector_type(8)));

#define HID 128
#define OUTC 10

// ---------------------------------------------------------------------------
// Edge scatter: agg[dst] += x[src].  One wave per edge, float4 per lane
// (32 lanes x 4 = 128 channels).  x/agg live in L2 (25.6 MB each).
// ---------------------------------------------------------------------------
__global__ void gin_scatter(const float* __restrict__ x,
                            const int* __restrict__ ei,
                            float* __restrict__ agg,
                            int n_edges) {
    long gid = (long)blockIdx.x * blockDim.x + threadIdx.x;
    int e    = (int)(gid >> 5);
    int lane = (int)(gid & 31);
    if (e >= n_edges) return;           // uniform per wave
    int src = ei[e];
    int dst = ei[n_edges + e];
    const float4* x4 = (const float4*)x;
    float4 v = x4[(long)src * 32 + lane];
    float* a = agg + (long)dst * HID + lane * 4;
    atomicAdd(a + 0, v.x);
    atomicAdd(a + 1, v.y);
    atomicAdd(a + 2, v.z);
    atomicAdd(a + 3, v.w);
}

// ---------------------------------------------------------------------------
// Fused GIN node MLP: xout = relu(relu((x+agg)@W1 + b1)@W2 + b2)
// One wave per 16-row tile.  Full-precision f32 WMMA 16x16x4.
//
// A-frag (16x4 f32): lane<16 holds (A[M=lane][k0], A[M=lane][k0+1]),
//                    lane>=16 holds (A[M=lane-16][k0+2], A[M=lane-16][k0+3]).
// So lane keeps its row's channel pairs at 4j+2*half -> 64 VGPRs for 16x128.
// B-frag (4x16 f32): lane n<16 -> (B[k0][n], B[k0+1][n]); lane>=16 -> k0+2/+3.
// D (16x16 f32): VGPR v, lane l<16 -> D[v][l]; lane>=16 -> D[v+8][l-16].
// ---------------------------------------------------------------------------
__global__ void __launch_bounds__(32)
gin_conv(const float* __restrict__ xin, const float* __restrict__ agg,
         const float* __restrict__ w1, const float* __restrict__ b1,
         const float* __restrict__ w2, const float* __restrict__ b2,
         float* __restrict__ xout, int n_nodes) {
    __shared__ float lds_h[16 * HID];           // 8 KB, wave-private
    const int tile  = blockIdx.x;
    const int lane  = threadIdx.x & 31;
    const int half  = lane >> 4;
    const int rlane = lane & 15;

    int row  = tile * 16 + rlane;
    int rowc = row < n_nodes ? row : n_nodes - 1;   // clamp loads; stores masked

    const float2* x2 = (const float2*)xin;
    const float2* g2 = (const float2*)agg;
    v2f a[32];
#pragma unroll
    for (int j = 0; j < 32; ++j) {
        float2 xv = x2[(long)rowc * 64 + 2 * j + half];
        float2 gv = g2[(long)rowc * 64 + 2 * j + half];
        a[j][0] = xv.x + gv.x;
        a[j][1] = xv.y + gv.y;
    }

    // ---- GEMM1: h = relu(A @ W1 + b1) -> LDS (D-layout -> [M][N]) ----
    for (int nt = 0; nt < 8; ++nt) {
        int col = nt * 16 + rlane;
        v8f acc = {};
#pragma unroll
        for (int j = 0; j < 32; ++j) {
            int k = 4 * j + 2 * half;
            v2f bf;
            bf[0] = w1[k * HID + col];
            bf[1] = w1[(k + 1) * HID + col];
            acc = __builtin_amdgcn_wmma_f32_16x16x4_f32(
                false, a[j], false, bf, (short)0, acc, false, false);
        }
        float bias = b1[col];
#pragma unroll
        for (int v = 0; v < 8; ++v) {
            float h = acc[v] + bias;
            lds_h[(v + 8 * half) * HID + col] = h > 0.f ? h : 0.f;
        }
    }
    __syncthreads();

    // reload intermediate tile in A-layout
    const float2* l2 = (const float2*)lds_h;
#pragma unroll
    for (int j = 0; j < 32; ++j) {
        float2 hv = l2[rlane * 64 + 2 * j + half];
        a[j][0] = hv.x;
        a[j][1] = hv.y;
    }

    // ---- GEMM2: xout = relu(H @ W2 + b2) ----
    for (int nt = 0; nt < 8; ++nt) {
        int col = nt * 16 + rlane;
        v8f acc = {};
#pragma unroll
        for (int j = 0; j < 32; ++j) {
            int k = 4 * j + 2 * half;
            v2f bf;
            bf[0] = w2[k * HID + col];
            bf[1] = w2[(k + 1) * HID + col];
            acc = __builtin_amdgcn_wmma_f32_16x16x4_f32(
                false, a[j], false, bf, (short)0, acc, false, false);
        }
        float bias = b2[col];
#pragma unroll
        for (int v = 0; v < 8; ++v) {
            int orow = tile * 16 + v + 8 * half;
            if (orow < n_nodes) {
                float h = acc[v] + bias;
                xout[(long)orow * HID + col] = h > 0.f ? h : 0.f;
            }
        }
    }
}

// ---------------------------------------------------------------------------
// Global add pool: g[batch[node]] += x[node].  One wave per node.
// ---------------------------------------------------------------------------
__global__ void gin_pool(const float* __restrict__ x,
                         const int* __restrict__ batch,
                         float* __restrict__ g, int n_nodes) {
    long gid = (long)blockIdx.x * blockDim.x + threadIdx.x;
    int node = (int)(gid >> 5);
    int lane = (int)(gid & 31);
    if (node >= n_nodes) return;
    int gr = batch[node];
    const float4* x4 = (const float4*)x;
    float4 v = x4[(long)node * 32 + lane];
    float* p = g + (long)gr * HID + lane * 4;
    atomicAdd(p + 0, v.x);
    atomicAdd(p + 1, v.y);
    atomicAdd(p + 2, v.z);
    atomicAdd(p + 3, v.w);
}

// ---------------------------------------------------------------------------
// MLP stage 1: h = relu(g @ W + b), one wave per 16-row tile (f32 WMMA).
// ---------------------------------------------------------------------------
__global__ void __launch_bounds__(32)
gin_mlp1(const float* __restrict__ g, const float* __restrict__ w,
         const float* __restrict__ b, float* __restrict__ hout, int n_rows) {
    const int tile  = blockIdx.x;
    const int lane  = threadIdx.x & 31;
    const int half  = lane >> 4;
    const int rlane = lane & 15;
    int row  = tile * 16 + rlane;
    int rowc = row < n_rows ? row : n_rows - 1;

    const float2* g2 = (const float2*)g;
    v2f a[32];
#pragma unroll
    for (int j = 0; j < 32; ++j) {
        float2 v = g2[(long)rowc * 64 + 2 * j + half];
        a[j][0] = v.x;
        a[j][1] = v.y;
    }
    for (int nt = 0; nt < 8; ++nt) {
        int col = nt * 16 + rlane;
        v8f acc = {};
#pragma unroll
        for (int j = 0; j < 32; ++j) {
            int k = 4 * j + 2 * half;
            v2f bf;
            bf[0] = w[k * HID + col];
            bf[1] = w[(k + 1) * HID + col];
            acc = __builtin_amdgcn_wmma_f32_16x16x4_f32(
                false, a[j], false, bf, (short)0, acc, false, false);
        }
        float bias = b[col];
#pragma unroll
        for (int v = 0; v < 8; ++v) {
            int orow = tile * 16 + v + 8 * half;
            if (orow < n_rows) {
                float h = acc[v] + bias;
                hout[(long)orow * HID + col] = h > 0.f ? h : 0.f;
            }
        }
    }
}

// ---------------------------------------------------------------------------
// MLP stage 2: out = h @ W2 + b2   (500x128x10 — trivial, scalar)
// ---------------------------------------------------------------------------
__global__ void gin_mlp2(const float* __restrict__ h, const float* __restrict__ w,
                         const float* __restrict__ b, float* __restrict__ out,
                         int n_rows) {
    int gid = blockIdx.x * blockDim.x + threadIdx.x;
    if (gid >= n_rows * OUTC) return;
    int r = gid / OUTC, o = gid % OUTC;
    float s = b[o];
#pragma unroll 8
    for (int k = 0; k < HID; ++k) s += h[(long)r * HID + k] * w[k * OUTC + o];
    out[gid] = s;
}

extern "C" void kernel_launch(void* const* d_in, const int* in_sizes, int n_in,
                              void* d_out, int out_size, void* d_ws, size_t ws_size,
                              hipStream_t stream) {
    const float* x   = (const float*)d_in[0];
    const float* cw1 = (const float*)d_in[1];
    const float* cb1 = (const float*)d_in[2];
    const float* cw2 = (const float*)d_in[3];
    const float* cb2 = (const float*)d_in[4];
    const float* mw1 = (const float*)d_in[5];
    const float* mb1 = (const float*)d_in[6];
    const float* mw2 = (const float*)d_in[7];
    const float* mb2 = (const float*)d_in[8];
    const int*   ei    = (const int*)d_in[9];
    const int*   batch = (const int*)d_in[10];

    const int n_nodes  = in_sizes[0] / HID;
    const int n_edges  = in_sizes[9] / 2;
    const int n_graphs = out_size / OUTC;

    // workspace partition
    float* agg   = (float*)d_ws;                         // n_nodes*HID
    float* bufA  = agg  + (size_t)n_nodes * HID;         // n_nodes*HID
    float* bufB  = bufA + (size_t)n_nodes * HID;         // n_nodes*HID
    float* gpool = bufB + (size_t)n_nodes * HID;         // n_graphs*HID
    float* hpool = gpool + (size_t)n_graphs * HID;       // n_graphs*HID

    const int ntiles = (n_nodes + 15) / 16;
    const float* xcur = x;
    float* pp[2] = {bufA, bufB};

    for (int l = 0; l < 3; ++l) {
        hipMemsetAsync(agg, 0, (size_t)n_nodes * HID * sizeof(float), stream);
        long tot = (long)n_edges * 32;
        gin_scatter<<<(int)((tot + 255) / 256), 256, 0, stream>>>(xcur, ei, agg, n_edges);
        float* xo = pp[l & 1];
        gin_conv<<<ntiles, 32, 0, stream>>>(
            xcur, agg,
            cw1 + (size_t)l * HID * HID, cb1 + (size_t)l * HID,
            cw2 + (size_t)l * HID * HID, cb2 + (size_t)l * HID,
            xo, n_nodes);
        xcur = xo;
    }

    hipMemsetAsync(gpool, 0, (size_t)n_graphs * HID * sizeof(float), stream);
    long totp = (long)n_nodes * 32;
    gin_pool<<<(int)((totp + 255) / 256), 256, 0, stream>>>(xcur, batch, gpool, n_nodes);

    int gtiles = (n_graphs + 15) / 16;
    gin_mlp1<<<gtiles, 32, 0, stream>>>(gpool, mw1, mb1, hpool, n_graphs);
    gin_mlp2<<<(n_graphs * OUTC + 255) / 256, 256, 0, stream>>>(hpool, mw2, mb2,
                                                                (float*)d_out, n_graphs);
}